// KNNSelector_49211735277876
// MI455X (gfx1250) — compile-verified
//
#include <hip/hip_runtime.h>

// Problem constants (match reference setup_inputs)
#define B_      4
#define G_      1024
#define N_      65536
#define K_      32
#define QT_     (G_/16)          // 64 query tiles of 16
#define NSPLIT_ 4                // split scene dimension across blocks
#define SEGN_   (N_/NSPLIT_)     // 16384 points per segment
#define CHUNK_  256              // points per block iteration (8 waves * 2 tiles * 16)
#define ITERS_  (SEGN_/CHUNK_)   // 64
#define QTOT_   (B_*G_)          // 4096 queries total
#define SROW_   260              // LDS row stride (floats): 8*260 % 64 == 32 -> conflict-free

typedef float v2f __attribute__((ext_vector_type(2)));
typedef float v8f __attribute__((ext_vector_type(8)));

// ---------------------------------------------------------------------------
// Kernel 1: per (batch, 16-query tile, scene segment) block, 256 threads.
//   - Two independent V_WMMA_F32_16X16X4_F32 per wave per iteration compute
//     q.p for two 16x16 tiles (A = queries x [x,y,z,0], B = [x,y,z,0] x pts).
//   - score s = ||p||^2 - 2 q.p (same per-query ordering as distance; scene
//     mask folded in as +inf so masked points can never win).
//   - 16 threads/query keep private sorted top-32 register lists fed by
//     ds_load_b64 pairs, then a half-wave shuffle multiway-merge emits the
//     segment top-32 (sorted ascending).
// ---------------------------------------------------------------------------
__global__ __launch_bounds__(256) void knn_main_kernel(
    const float* __restrict__ grasp,   // (B,G,3)
    const float* __restrict__ xyz,     // (B,N,3)
    const int*   __restrict__ smask,   // (B,N)
    float* __restrict__ wsd,           // (QTOT, NSPLIT, 32) segment top dists
    int*   __restrict__ wsi)           // (QTOT, NSPLIT, 32) segment top idxs
{
    const float FINF = __builtin_inff();
    __shared__ float sdist[16 * SROW_];  // 16 queries x 256 cols (+pad)

    const int bx    = blockIdx.x;
    const int seg   = bx % NSPLIT_;
    const int tmp   = bx / NSPLIT_;
    const int qtile = tmp % QT_;
    const int b     = tmp / QT_;

    const int t    = threadIdx.x;
    const int w    = t >> 5;          // wave 0..7
    const int lane = t & 31;
    const int g    = lane & 15;       // position in half-wave / query group
    const int half = lane >> 4;       // which half of the wave
    const int q    = t >> 4;          // query 0..15 this thread scans/merges

    // --- A fragment: 16 queries x K=4 (x,y,z,0), f32 WMMA A layout:
    // lanes 0-15 hold K=0 (vgpr0) / K=1 (vgpr1); lanes 16-31 hold K=2 / K=3.
    const int qglob = qtile * 16 + g;
    const float* qp = grasp + ((size_t)b * G_ + qglob) * 3;
    const float qx = qp[0], qy = qp[1], qz = qp[2];
    v2f afrag;
    afrag.x = half ? qz : qx;
    afrag.y = half ? 0.0f : qy;

    // --- private top-K list, sorted ascending (worst at [K-1])
    float ld[K_]; int li[K_];
#pragma unroll
    for (int j = 0; j < K_; ++j) { ld[j] = FINF; li[j] = 0x7fffffff; }

    const int segbase = seg * SEGN_;
    for (int it = 0; it < ITERS_; ++it) {
        const int pchunk = segbase + it * CHUNK_;

        // --- two point tiles per wave: cols [w*16, w*16+16) and +128
        const int j0 = pchunk + w * 16 + g;
        const int j1 = j0 + 128;
        const float* pp0 = xyz + ((size_t)b * N_ + j0) * 3;
        const float* pp1 = xyz + ((size_t)b * N_ + j1) * 3;
        const float p0x = pp0[0], p0y = pp0[1], p0z = pp0[2];
        const float p1x = pp1[0], p1y = pp1[1], p1z = pp1[2];
        const int   mk0 = smask[(size_t)b * N_ + j0];
        const int   mk1 = smask[(size_t)b * N_ + j1];
        const float p20 = fmaf(p0x, p0x, fmaf(p0y, p0y, p0z * p0z));
        const float p21 = fmaf(p1x, p1x, fmaf(p1y, p1y, p1z * p1z));
        const float p2m0 = mk0 ? p20 : FINF;   // mask folded into ||p||^2
        const float p2m1 = mk1 ? p21 : FINF;
        v2f bf0, bf1;
        bf0.x = half ? p0z : p0x;  bf0.y = half ? 0.0f : p0y;
        bf1.x = half ? p1z : p1x;  bf1.y = half ? 0.0f : p1y;

        v8f c0 = {}, c1 = {};
        c0 = __builtin_amdgcn_wmma_f32_16x16x4_f32(
                 false, afrag, false, bf0, (short)0, c0, false, false);
        c1 = __builtin_amdgcn_wmma_f32_16x16x4_f32(
                 false, afrag, false, bf1, (short)0, c1, false, false);

        // C layout: lane holds column n = lane&15, rows m = v + 8*half.
        const int colbase = w * 16 + g;
        const int rowoff  = half * 8;
#pragma unroll
        for (int v = 0; v < 8; ++v) {
            sdist[(v + rowoff) * SROW_ + colbase]       = fmaf(-2.0f, c0[v], p2m0);
            sdist[(v + rowoff) * SROW_ + colbase + 128] = fmaf(-2.0f, c1[v], p2m1);
        }
        __syncthreads();

        // --- scan: thread (q,g) takes column pairs {2g,2g+1} + 32*cc
        //     (ds_load_b64, increasing index order)
#pragma unroll
        for (int cc = 0; cc < 8; ++cc) {
            const int col = cc * 32 + 2 * g;
            const float2 sv = *(const float2*)&sdist[q * SROW_ + col];
#pragma unroll
            for (int e = 0; e < 2; ++e) {
                const float s   = (e == 0) ? sv.x : sv.y;
                const int   idx = pchunk + col + e;
                if (s < ld[K_ - 1]) {               // beats current worst?
                    ld[K_ - 1] = s; li[K_ - 1] = idx;
#pragma unroll
                    for (int jj = K_ - 1; jj > 0; --jj) {  // one bubble pass
                        if (ld[jj] < ld[jj - 1]) {
                            float td = ld[jj]; ld[jj] = ld[jj - 1]; ld[jj - 1] = td;
                            int   ti = li[jj]; li[jj] = li[jj - 1]; li[jj - 1] = ti;
                        }
                    }
                }
            }
        }
        __syncthreads();
    }

    // --- half-wave multiway merge: 16 sorted lists -> segment top-32 (sorted)
    const int Q = (b * QT_ + qtile) * 16 + q;
    const size_t outb = ((size_t)Q * NSPLIT_ + seg) * K_;
    for (int k = 0; k < K_; ++k) {
        float bv = ld[0]; int bi = li[0];
#pragma unroll
        for (int m = 8; m >= 1; m >>= 1) {      // width-16 min-reduce
            const float ov = __shfl_xor(bv, m, 16);
            const int   oi = __shfl_xor(bi, m, 16);
            if (ov < bv || (ov == bv && oi < bi)) { bv = ov; bi = oi; }
        }
        if (g == 0) { wsd[outb + k] = bv; wsi[outb + k] = bi; }
        if (ld[0] == bv && li[0] == bi) {       // winner pops its head
#pragma unroll
            for (int jj = 0; jj < K_ - 1; ++jj) { ld[jj] = ld[jj + 1]; li[jj] = li[jj + 1]; }
            ld[K_ - 1] = FINF; li[K_ - 1] = 0x7fffffff;
        }
    }
}

// ---------------------------------------------------------------------------
// Kernel 2: one wave per query merges NSPLIT*32 = 128 candidates -> final 32.
// ---------------------------------------------------------------------------
__global__ __launch_bounds__(256) void knn_merge_kernel(
    const float* __restrict__ wsd, const int* __restrict__ wsi,
    int* __restrict__ oIdx, float* __restrict__ oMask)
{
    const float FINF = __builtin_inff();
    const int Q    = blockIdx.x * 8 + (threadIdx.x >> 5);
    const int lane = threadIdx.x & 31;

    float dv[NSPLIT_]; int iv[NSPLIT_];
    const size_t base = (size_t)Q * NSPLIT_ * K_;
#pragma unroll
    for (int s = 0; s < NSPLIT_; ++s) {
        dv[s] = wsd[base + s * K_ + lane];
        iv[s] = wsi[base + s * K_ + lane];
    }

    for (int k = 0; k < K_; ++k) {
        float bv = dv[0]; int bi = iv[0];
#pragma unroll
        for (int s = 1; s < NSPLIT_; ++s)
            if (dv[s] < bv || (dv[s] == bv && iv[s] < bi)) { bv = dv[s]; bi = iv[s]; }
#pragma unroll
        for (int m = 16; m >= 1; m >>= 1) {     // wave32 min-reduce
            const float ov = __shfl_xor(bv, m, 32);
            const int   oi = __shfl_xor(bi, m, 32);
            if (ov < bv || (ov == bv && oi < bi)) { bv = ov; bi = oi; }
        }
        if (lane == 0) {
            oIdx[Q * K_ + k]  = bi;
            oMask[Q * K_ + k] = (bv < FINF) ? 1.0f : 0.0f;
        }
#pragma unroll
        for (int s = 0; s < NSPLIT_; ++s)       // invalidate the winner
            if (dv[s] == bv && iv[s] == bi) { dv[s] = FINF; iv[s] = 0x7fffffff; }
    }
}

extern "C" void kernel_launch(void* const* d_in, const int* in_sizes, int n_in,
                              void* d_out, int out_size, void* d_ws, size_t ws_size,
                              hipStream_t stream) {
    (void)in_sizes; (void)n_in; (void)out_size; (void)ws_size;
    const float* grasp = (const float*)d_in[0];   // (B,G,3) f32
    const float* xyz   = (const float*)d_in[1];   // (B,N,3) f32
    const int*   smask = (const int*)d_in[2];     // (B,N)   i32

    // workspace: (QTOT * NSPLIT * 32) dists then same count of indices (4 MB)
    float* wsd = (float*)d_ws;
    int*   wsi = (int*)((char*)d_ws + (size_t)QTOT_ * NSPLIT_ * K_ * sizeof(float));

    // output: indices (int32, QTOT*K) then local_mask (float, QTOT*K)
    int*   oIdx  = (int*)d_out;
    float* oMask = (float*)d_out + (size_t)QTOT_ * K_;

    knn_main_kernel<<<dim3(B_ * QT_ * NSPLIT_), dim3(256), 0, stream>>>(
        grasp, xyz, smask, wsd, wsi);
    knn_merge_kernel<<<dim3(QTOT_ / 8), dim3(256), 0, stream>>>(
        wsd, wsi, oIdx, oMask);
}